// GATLayer_10952166605248
// MI455X (gfx1250) — compile-verified
//
#include <hip/hip_runtime.h>
#include <hip/hip_bf16.h>

typedef __attribute__((ext_vector_type(2))) float v2f;
typedef __attribute__((ext_vector_type(8))) float v8f;

#define LEAKY_SLOPE 0.05f

// Problem geometry (fixed by the harness): F=256, H=64, N=50000 (=3125*16), E=1.6M
#define FDIM 256
#define HDIM 64
#define BSTRIDE 268   // 256 + 12: bank-conflict-free fragment reads, 16B-aligned rows

// ---------------------------------------------------------------------------
// Kernel 1: zero out[] (N*H) and h_sum[] (N)
// ---------------------------------------------------------------------------
__global__ void zero_kernel(float* __restrict__ out, float* __restrict__ h_sum,
                            int nOut, int nSum) {
    int i = blockIdx.x * blockDim.x + threadIdx.x;
    if (i < nOut) out[i] = 0.0f;
    if (i < nSum) h_sum[i] = 0.0f;
}

// ---------------------------------------------------------------------------
// Kernel 2: z = x @ fc_w^T via V_WMMA_F32_16X16X4_F32
//   x: [N, FDIM] row-major, fc_w: [HDIM, FDIM] row-major, z: [N, HDIM]
// fc_w is staged in LDS (padded stride) once per block; one wave computes a
// 16-row tile across all 64 output columns (4 n-tiles, compile-time unrolled).
//
// A-frag (16x4 f32): lane L<16  -> v0=A[M=L][k0+0],   v1=A[M=L][k0+1]
//                    lane L>=16 -> v0=A[M=L-16][k0+2], v1=A[M=L-16][k0+3]
// B-frag (4x16 f32): lane L<16  -> v0=B[k0+0][N=L],    v1=B[k0+1][N=L]
//                    lane L>=16 -> v0=B[k0+2][N=L-16], v1=B[k0+3][N=L-16]
//                    (B = fc_w^T, so B[k][n] = fc_w[n][k]: contiguous pairs)
// C/D (16x16 f32):   vgpr r, lane L -> M = r + 8*(L>>4), Ncol = L&15
// ---------------------------------------------------------------------------
__global__ void gemm_z_wmma(const float* __restrict__ x,
                            const float* __restrict__ w,
                            float* __restrict__ z,
                            int N) {
    __shared__ float Bs[HDIM * BSTRIDE];   // ~67 KB

    // ---- stage fc_w into LDS (all threads, incl. tail waves) ----
    {
        const int groupsPerRow = FDIM >> 2;                 // 64 float4 per row
        for (int g = threadIdx.x; g < HDIM * groupsPerRow; g += blockDim.x) {
            int row = g >> 6;                               // g / 64
            int c4  = g & 63;                               // g % 64
            float4 v = *(const float4*)(w + (size_t)row * FDIM + (c4 << 2));
            *(float4*)(&Bs[row * BSTRIDE + (c4 << 2)]) = v;
        }
    }
    __syncthreads();

    const int lane   = threadIdx.x & 31;
    const int wave   = threadIdx.x >> 5;
    const int mTile  = blockIdx.x * (blockDim.x >> 5) + wave;
    const int mTiles = N >> 4;                              // N multiple of 16
    if (mTile >= mTiles) return;                            // wave-uniform exit
    const int m0    = mTile << 4;
    const int laneM = lane & 15;
    const int kOff  = (lane >> 4) << 1;                     // 0 or 2

    v8f acc0 = {0,0,0,0,0,0,0,0};
    v8f acc1 = acc0, acc2 = acc0, acc3 = acc0;

    const float* xrow = x + (size_t)(m0 + laneM) * FDIM + kOff;
    const float* b0 = &Bs[( 0 + laneM) * BSTRIDE + kOff];
    const float* b1 = &Bs[(16 + laneM) * BSTRIDE + kOff];
    const float* b2 = &Bs[(32 + laneM) * BSTRIDE + kOff];
    const float* b3 = &Bs[(48 + laneM) * BSTRIDE + kOff];

#pragma unroll 4
    for (int k0 = 0; k0 < FDIM; k0 += 4) {
        v2f a   = *(const v2f*)(xrow + k0);
        v2f vb0 = *(const v2f*)(b0 + k0);
        v2f vb1 = *(const v2f*)(b1 + k0);
        v2f vb2 = *(const v2f*)(b2 + k0);
        v2f vb3 = *(const v2f*)(b3 + k0);
        acc0 = __builtin_amdgcn_wmma_f32_16x16x4_f32(false, a, false, vb0,
                                                     (short)0, acc0, false, false);
        acc1 = __builtin_amdgcn_wmma_f32_16x16x4_f32(false, a, false, vb1,
                                                     (short)0, acc1, false, false);
        acc2 = __builtin_amdgcn_wmma_f32_16x16x4_f32(false, a, false, vb2,
                                                     (short)0, acc2, false, false);
        acc3 = __builtin_amdgcn_wmma_f32_16x16x4_f32(false, a, false, vb3,
                                                     (short)0, acc3, false, false);
    }

    // Store C/D: lane L, vgpr r holds z[m0 + r + 8*(L>>4)][t*16 + (L&15)]
    const int mhi = (lane >> 4) << 3;                       // 0 or 8
    float* zrow = z + (size_t)(m0 + mhi) * HDIM + laneM;
#pragma unroll
    for (int r = 0; r < 8; ++r) {
        float* zr = zrow + (size_t)r * HDIM;
        zr[ 0] = acc0[r];
        zr[16] = acc1[r];
        zr[32] = acc2[r];
        zr[48] = acc3[r];
    }
}

// ---------------------------------------------------------------------------
// Kernel 3: per-node attention scalars
//   s_src[n] = z[n] . a_w[0:H],  s_dst[n] = z[n] . a_w[H:2H]
// ---------------------------------------------------------------------------
__global__ void dots_kernel(const float* __restrict__ z,
                            const float* __restrict__ a_w,
                            float* __restrict__ s_src,
                            float* __restrict__ s_dst,
                            int N) {
    int n = blockIdx.x * blockDim.x + threadIdx.x;
    if (n >= N) return;
    const float* zr = z + (size_t)n * HDIM;
    float ss = 0.0f, sd = 0.0f;
#pragma unroll
    for (int i = 0; i < HDIM; i += 4) {
        float4 v  = *(const float4*)(zr + i);
        float4 w0 = *(const float4*)(a_w + i);
        float4 w1 = *(const float4*)(a_w + HDIM + i);
        ss += v.x * w0.x + v.y * w0.y + v.z * w0.z + v.w * w0.w;
        sd += v.x * w1.x + v.y * w1.y + v.z * w1.z + v.w * w1.w;
    }
    s_src[n] = ss;
    s_dst[n] = sd;
}

// ---------------------------------------------------------------------------
// Kernel 4: per-edge numerator h = exp(leaky_relu(e)) + segment sum into
// h_sum[src] (f32 atomic; h_sum is 200 KB -> lives in L2)
// ---------------------------------------------------------------------------
__global__ void edge_h_kernel(const int* __restrict__ ei,
                              const float* __restrict__ s_src,
                              const float* __restrict__ s_dst,
                              const float* __restrict__ a_b,
                              float* __restrict__ h,
                              float* __restrict__ h_sum,
                              int E) {
    int e = blockIdx.x * blockDim.x + threadIdx.x;
    if (e >= E) return;
    int s = ei[e];
    int d = ei[E + e];
    float v = s_src[s] + s_dst[d] + a_b[0];
    v = (v > 0.0f) ? v : LEAKY_SLOPE * v;
    float hv = __expf(v);
    h[e] = hv;
    __hip_atomic_fetch_add(&h_sum[s], hv, __ATOMIC_RELAXED, __HIP_MEMORY_SCOPE_AGENT);
}

// ---------------------------------------------------------------------------
// Kernel 5: alpha[e] = h[e] / h_sum[src[e]]  (2nd output)
// ---------------------------------------------------------------------------
__global__ void alpha_kernel(const int* __restrict__ ei,
                             const float* __restrict__ h,
                             const float* __restrict__ h_sum,
                             float* __restrict__ alpha,
                             int E) {
    int e = blockIdx.x * blockDim.x + threadIdx.x;
    if (e >= E) return;
    alpha[e] = h[e] / h_sum[ei[e]];
}

// ---------------------------------------------------------------------------
// Kernel 6: out[src] += alpha * z[dst]
// One thread per (edge, quad-of-4-cols): float4 gather + 4 f32 atomics.
// z and out both fit in the 192 MB L2, so this is L2-atomic traffic.
// ---------------------------------------------------------------------------
__global__ void scatter_kernel(const int* __restrict__ ei,
                               const float* __restrict__ alpha,
                               const float* __restrict__ z,
                               float* __restrict__ out,
                               int E) {
    long long idx = (long long)blockIdx.x * blockDim.x + threadIdx.x;
    int e = (int)(idx >> 4);            // HDIM/4 = 16 quads per edge
    int q = (int)(idx & 15);
    if (e >= E) return;
    int s = ei[e];
    int d = ei[E + e];
    float a = alpha[e];
    float4 zv = *(const float4*)(z + (size_t)d * HDIM + (q << 2));
    float* o = out + (size_t)s * HDIM + (q << 2);
    __hip_atomic_fetch_add(o + 0, a * zv.x, __ATOMIC_RELAXED, __HIP_MEMORY_SCOPE_AGENT);
    __hip_atomic_fetch_add(o + 1, a * zv.y, __ATOMIC_RELAXED, __HIP_MEMORY_SCOPE_AGENT);
    __hip_atomic_fetch_add(o + 2, a * zv.z, __ATOMIC_RELAXED, __HIP_MEMORY_SCOPE_AGENT);
    __hip_atomic_fetch_add(o + 3, a * zv.w, __ATOMIC_RELAXED, __HIP_MEMORY_SCOPE_AGENT);
}

// ---------------------------------------------------------------------------
// Launcher
// Inputs:  d_in[0]=x [N,F] f32, d_in[1]=edge_index [2,E] int, d_in[2]=fc_w [H,F] f32,
//          d_in[3]=a_w [2H] f32, d_in[4]=a_b [1] f32
// Output:  d_out = out [N,H] f32 ++ alpha [E] f32 (concatenated flat)
// Scratch: z [N*H] | s_src [N] | s_dst [N] | h_sum [N] | h [E]  (~19.8 MB)
// ---------------------------------------------------------------------------
extern "C" void kernel_launch(void* const* d_in, const int* in_sizes, int n_in,
                              void* d_out, int out_size, void* d_ws, size_t ws_size,
                              hipStream_t stream) {
    const float* x    = (const float*)d_in[0];
    const int*   ei   = (const int*)d_in[1];
    const float* fc_w = (const float*)d_in[2];
    const float* a_w  = (const float*)d_in[3];
    const float* a_b  = (const float*)d_in[4];

    const int N = in_sizes[0] / FDIM;     // 50000
    const int E = in_sizes[1] / 2;        // 1600000

    float* out   = (float*)d_out;
    float* alpha = out + (size_t)N * HDIM;

    float* z     = (float*)d_ws;
    float* s_src = z + (size_t)N * HDIM;
    float* s_dst = s_src + N;
    float* h_sum = s_dst + N;
    float* h     = h_sum + N;

    // 1. zero out[] and h_sum[]
    {
        int nOut = N * HDIM;
        zero_kernel<<<(nOut + 255) / 256, 256, 0, stream>>>(out, h_sum, nOut, N);
    }
    // 2. z = x @ fc_w^T (WMMA, fc_w staged in LDS)
    {
        int mTiles = N >> 4;
        int wavesPerBlock = 8;                               // 256 threads
        int blocks = (mTiles + wavesPerBlock - 1) / wavesPerBlock;
        gemm_z_wmma<<<blocks, wavesPerBlock * 32, 0, stream>>>(x, fc_w, z, N);
    }
    // 3. per-node attention scalars
    dots_kernel<<<(N + 255) / 256, 256, 0, stream>>>(z, a_w, s_src, s_dst, N);
    // 4. per-edge h + segment sum
    edge_h_kernel<<<(E + 255) / 256, 256, 0, stream>>>(ei, s_src, s_dst, a_b, h, h_sum, E);
    // 5. alpha
    alpha_kernel<<<(E + 255) / 256, 256, 0, stream>>>(ei, h, h_sum, alpha, E);
    // 6. scatter out[src] += alpha * z[dst]
    {
        long long total = (long long)E * (HDIM / 4);         // 25.6M threads
        int blocks = (int)((total + 255) / 256);
        scatter_kernel<<<blocks, 256, 0, stream>>>(ei, alpha, z, out, E);
    }
}